// CVCalculator_82583631167654
// MI455X (gfx1250) — compile-verified
//
#include <hip/hip_runtime.h>
#include <cstdint>
#include <cstddef>

// Sliding-window (W=60) coefficient of variation over (B=16, T=4096, F=64) fp32.
// Memory-bound: ~32MB HBM traffic -> ~1.4us floor at 23.3 TB/s.
// gfx1250 path: async global->LDS DMA (ASYNCcnt) staging + s_wait_asynccnt fence,
// LDS-resident running-window scan, nontemporal output stores.

namespace {
constexpr int kB    = 16;
constexpr int kT    = 4096;
constexpr int kF    = 64;
constexpr int kWin  = 60;
constexpr int kHalo = kWin - 1;            // 59
constexpr float kEps = 1e-6f;

constexpr int kChunk   = 128;              // output timesteps per block
constexpr int kSub     = 32;               // output timesteps per thread
constexpr int kYDim    = kChunk / kSub;    // 4 -> 256 threads = 8 wave32
constexpr int kThreads = kF * kYDim;
constexpr int kMaxRows = kChunk + kHalo;   // 187 rows staged (47,872 B LDS)
} // namespace

__global__ __launch_bounds__(kThreads)
void cv_sliding_kernel(const float* __restrict__ flow, float* __restrict__ out) {
  __shared__ float smem[kMaxRows * kF];    // row-major: [row][feature], bank = feature

  const int chunk    = blockIdx.x;
  const int b        = blockIdx.y;
  const int t0       = chunk * kChunk;
  const int in_start = (t0 > kHalo) ? (t0 - kHalo) : 0;   // first staged global row
  const int n_rows   = t0 + kChunk - in_start;            // 187 (128 for chunk 0)

  // ---- Stage rows [in_start, in_start + n_rows) of batch b into LDS via
  //      gfx1250 async DMA: one 128-bit packet per lane per issue. ----
  const int tid     = threadIdx.x + threadIdx.y * kF;
  const int total16 = n_rows * ((kF * 4) / 16);           // # of 16-byte packets
  const uint64_t gbase =
      (uint64_t)(uintptr_t)(flow + ((size_t)b * kT + in_start) * kF);
  const uint32_t lbase = (uint32_t)(uintptr_t)(&smem[0]); // LDS byte offset = addr[31:0]

  for (int i = tid; i < total16; i += kThreads) {
    uint32_t laddr = lbase + (uint32_t)i * 16u;
    uint64_t gaddr = gbase + (uint64_t)i * 16u;
    asm volatile("global_load_async_to_lds_b128 %0, %1, off"
                 :
                 : "v"(laddr), "v"(gaddr)
                 : "memory");
  }
  // Each wave drains its own async DMA, then the workgroup barrier makes all
  // staged rows visible to every wave.
  asm volatile("s_wait_asynccnt 0" ::: "memory");
  __syncthreads();

  // ---- Running-window scan out of LDS. Thread = (feature f, time subchunk). ----
  const int f  = threadIdx.x;
  const int ts = t0 + threadIdx.y * kSub;                 // first output timestep
  int left = (ts > kHalo) ? (ts - kHalo) : 0;             // left edge of window

  float s1 = 0.0f, s2 = 0.0f;
  // Warm-up: accumulate rows [left, ts) (<= 59 rows, all staged in LDS).
  for (int t = left; t < ts; ++t) {
    float x = smem[(t - in_start) * kF + f];
    s1 += x;
    s2 += x * x;
  }

  const float inv_w  = 1.0f / (float)kWin;
  const float inv_w1 = 1.0f / (float)(kWin - 1);

  for (int t = ts; t < ts + kSub; ++t) {
    float x = smem[(t - in_start) * kF + f];
    s1 += x;
    s2 += x * x;
    if (t - left + 1 > kWin) {                            // evict oldest row
      float y = smem[(left - in_start) * kF + f];
      s1 -= y;
      s2 -= y * y;
      ++left;
    }
    float res = 0.0f;                                     // zero pad for t < 59
    if (t >= kHalo) {
      float mean = s1 * inv_w;
      float var  = (s2 - s1 * mean) * inv_w1;             // unbiased (ddof=1)
      var = var > 0.0f ? var : 0.0f;
      res = sqrtf(var) / (mean + kEps);
    }
    // Output is write-once, never re-read: bypass cache retention (TH=NT).
    __builtin_nontemporal_store(res, &out[((size_t)b * kT + t) * kF + f]);
  }
}

extern "C" void kernel_launch(void* const* d_in, const int* in_sizes, int n_in,
                              void* d_out, int out_size, void* d_ws, size_t ws_size,
                              hipStream_t stream) {
  (void)in_sizes; (void)n_in; (void)out_size; (void)d_ws; (void)ws_size;
  const float* flow = (const float*)d_in[0];
  float* out        = (float*)d_out;
  dim3 grid(kT / kChunk, kB);      // (32, 16) = 512 blocks
  dim3 block(kF, kYDim);           // (64, 4)  = 256 threads = 8 wave32
  hipLaunchKernelGGL(cv_sliding_kernel, grid, block, 0, stream, flow, out);
}